// LinearDecisionTransformerDecoupled_51333449122143
// MI455X (gfx1250) — compile-verified
//
#include <hip/hip_runtime.h>
#include <math.h>

// ---- problem constants (match reference) ----
#define BB     2
#define KK     512
#define SDIM   60
#define NACT   5
#define EE     256
#define NHEAD  4
#define HDIM   64
#define FFD    1024
#define NLAYER 2
#define LSEQ   (3 * KK)      // 1536 tokens
#define MROWS  (BB * LSEQ)   // 3072 rows
#define EPS_F  1e-6f
#define LN_EPS 1e-5f
#define CHUNK  8
#define NCHUNK (LSEQ / CHUNK)   // 192

typedef float v2f __attribute__((ext_vector_type(2)));
typedef float v8f __attribute__((ext_vector_type(8)));

__device__ __forceinline__ float wave_sum32(float v) {
  #pragma unroll
  for (int off = 16; off > 0; off >>= 1) v += __shfl_xor(v, off, 32);
  return v;
}

// ---------------------------------------------------------------------------
// Token embedding: interleave (rtg, state, action) projections + pos_emb.
// ---------------------------------------------------------------------------
__global__ void embed_kernel(const float* __restrict__ rtg,
                             const float* __restrict__ state,
                             const float* __restrict__ action,
                             const float* __restrict__ rtg_w,  const float* __restrict__ rtg_b,
                             const float* __restrict__ state_w,const float* __restrict__ state_b,
                             const float* __restrict__ action_w,const float* __restrict__ action_b,
                             const float* __restrict__ pos,
                             float* __restrict__ x) {
  const int row = blockIdx.x;           // 0 .. MROWS-1
  const int e   = threadIdx.x;          // 0 .. 255
  const int b   = row / LSEQ;
  const int rr  = row % LSEQ;
  const int k   = rr / 3;
  const int t   = rr % 3;
  float val;
  if (t == 0) {
    val = rtg[b * KK + k] * rtg_w[e] + rtg_b[e];
  } else if (t == 1) {
    val = state_b[e];
    const float* sp = state + (b * KK + k) * SDIM;
    #pragma unroll 4
    for (int j = 0; j < SDIM; ++j) val += sp[j] * state_w[j * EE + e];
  } else {
    val = action_b[e];
    const float* ap = action + (b * KK + k) * NACT;
    #pragma unroll
    for (int j = 0; j < NACT; ++j) val += ap[j] * action_w[j * EE + e];
  }
  x[row * EE + e] = val + pos[k * EE + e];
}

// ---------------------------------------------------------------------------
// LayerNorm over E=256: one wave32 per row, 8 elements per lane.
// ---------------------------------------------------------------------------
__global__ void layernorm_kernel(const float* __restrict__ x,
                                 const float* __restrict__ g,
                                 const float* __restrict__ b,
                                 float* __restrict__ y) {
  const int row  = blockIdx.x;
  const int lane = threadIdx.x;         // 0..31
  const float* xr = x + row * EE;
  float v[8];
  float s = 0.f;
  #pragma unroll
  for (int i = 0; i < 8; ++i) { v[i] = xr[lane + i * 32]; s += v[i]; }
  const float mean = wave_sum32(s) * (1.0f / EE);
  float var = 0.f;
  #pragma unroll
  for (int i = 0; i < 8; ++i) { float d = v[i] - mean; var += d * d; }
  var = wave_sum32(var) * (1.0f / EE);
  const float inv = rsqrtf(var + LN_EPS);
  float* yr = y + row * EE;
  #pragma unroll
  for (int i = 0; i < 8; ++i) {
    int e = lane + i * 32;
    yr[e] = (v[i] - mean) * inv * g[e] + b[e];
  }
}

// ---------------------------------------------------------------------------
// FP32 WMMA GEMM: C[M,N] = act(A[M,Kd] @ W[Kd,N] + bias) (+ res)
// One wave computes a 16 x (16*NT) strip with NT v8f accumulators, using
// V_WMMA_F32_16X16X4_F32 (native fp32 matrix path on CDNA5, wave32).
// M%16==0, N%(16*NT)==0, Kd%4==0; no divergence so EXEC stays all-ones.
// ---------------------------------------------------------------------------
template<int ACT, bool RES, int NT>
__global__ void gemm_wmma_f32(const float* __restrict__ A, const float* __restrict__ W,
                              const float* __restrict__ bias, const float* __restrict__ res,
                              float* __restrict__ C, int M, int Kd, int N) {
  const int wid     = (blockIdx.x * blockDim.x + threadIdx.x) >> 5;
  const int lane    = threadIdx.x & 31;
  const int tiles_n = N / (16 * NT);
  const int m0 = (wid / tiles_n) << 4;
  const int n0 = (wid % tiles_n) * (16 * NT);
  const int lm = lane & 15;
  const int kg = (lane >> 4) << 1;      // 0 (lanes 0-15) or 2 (lanes 16-31)
  const float* Ap = A + (m0 + lm) * Kd;
  v8f acc[NT];
  #pragma unroll
  for (int j = 0; j < NT; ++j) acc[j] = (v8f){};
  for (int k0 = 0; k0 < Kd; k0 += 4) {
    v2f a;                               // A 16x4: lane = M row, vgprs hold K
    a.x = Ap[k0 + kg];
    a.y = Ap[k0 + kg + 1];
    const float* Wp = W + (k0 + kg) * N + n0 + lm;  // B 4x16: lane = N col
    #pragma unroll
    for (int j = 0; j < NT; ++j) {
      v2f bf;
      bf.x = Wp[j * 16];
      bf.y = Wp[j * 16 + N];
      acc[j] = __builtin_amdgcn_wmma_f32_16x16x4_f32(false, a, false, bf,
                                                     (short)0, acc[j], false, false);
    }
  }
  // C/D layout: vgpr i -> M = m0 + i + 8*(lane>=16); N = n0 + j*16 + lane%16
  const int cg = (lane >> 4) << 3;
  #pragma unroll
  for (int j = 0; j < NT; ++j) {
    const int col = n0 + j * 16 + lm;
    const float bv = bias[col];
    #pragma unroll
    for (int i = 0; i < 8; ++i) {
      const int r = m0 + cg + i;
      float v = acc[j][i] + bv;
      if (ACT == 1) v = 0.5f * v * (1.0f + erff(v * 0.70710678118654752f)); // exact GELU
      if (RES) v += res[r * N + col];
      C[r * N + col] = v;
    }
  }
}

// ---------------------------------------------------------------------------
// Linear attention causal scan. One 256-thread block per (b, head).
// Thread layout: wave w owns output columns e in [8w, 8w+8); within a wave,
// lane = (dg, e): dg = lane>>3 owns d in [16*dg, 16*dg+16), e = 8w + (lane&7).
// S[d][e] and a wave-replicated z[d] live in registers, so the per-step
// q.S / q.z reductions are two shfl_xor butterflies (masks 8,16): no barriers
// inside a step. q/k/v are staged 8 timesteps at a time, double-buffered in
// LDS via CDNA5 async-tensor loads (global_load_async_to_lds_b32, ASYNCcnt):
// chunk c+1 is prefetched before chunk c is consumed (s_wait_asynccnt 6).
// ---------------------------------------------------------------------------
__global__ void linear_attn_scan(const float* __restrict__ qkv,
                                 float* __restrict__ attn) {
  const int bh   = blockIdx.x;          // 0 .. B*H-1
  const int b    = bh / NHEAD;
  const int h    = bh % NHEAD;
  const int tid  = threadIdx.x;         // 0..255
  const int lane = tid & 31;
  const int wave = tid >> 5;            // 0..7
  const int e    = (wave << 3) | (lane & 7);  // 0..63
  const int dg   = lane >> 3;           // 0..3
  const int d0   = dg << 4;

  // [buf][sel(q,k,v)][step][d]
  __shared__ float sbuf[2][3][CHUNK][HDIM];

  float S[16], Z[16];
  #pragma unroll
  for (int i = 0; i < 16; ++i) { S[i] = 0.f; Z[i] = 0.f; }

  // stage one chunk: 1536 floats, 6 async-to-LDS issues per wave
  auto stage = [&](int buf, int l0) {
    #pragma unroll
    for (int j = 0; j < 6; ++j) {
      const int idx = j * 256 + tid;        // 0..1535
      const int s   = idx / 192;            // timestep in chunk
      const int r   = idx % 192;
      const int sel = r >> 6;               // 0=q, 1=k, 2=v
      const int d   = r & 63;
      const float* gp = qkv + (size_t)(b * LSEQ + l0 + s) * (3 * EE)
                            + sel * EE + h * HDIM + d;
      const float* lp = &sbuf[buf][sel][s][d];
      asm volatile("global_load_async_to_lds_b32 %0, %1, off"
                   :: "v"((unsigned)(unsigned long long)lp),
                      "v"((unsigned long long)gp)
                   : "memory");
    }
  };

  stage(0, 0);
  for (int c = 0; c < NCHUNK; ++c) {
    if (c + 1 < NCHUNK) {
      stage((c + 1) & 1, (c + 1) * CHUNK);  // overlap next chunk's DMA
      asm volatile("s_wait_asynccnt 0x6" ::: "memory");  // chunk c landed
    } else {
      asm volatile("s_wait_asynccnt 0x0" ::: "memory");
    }
    __syncthreads();
    const int buf = c & 1;
    #pragma unroll
    for (int s = 0; s < CHUNK; ++s) {
      const float ve = sbuf[buf][2][s][e];
      float p = 0.f, dp = 0.f;
      #pragma unroll
      for (int i = 0; i < 16; ++i) {
        const float kd = fmaxf(sbuf[buf][1][s][d0 + i], 0.f) + EPS_F;
        const float qd = fmaxf(sbuf[buf][0][s][d0 + i], 0.f) + EPS_F;
        S[i] += kd * ve;                 // rank-1 state update (inclusive)
        Z[i] += kd;                      // wave-replicated k-cumsum
        p    += qd * S[i];
        dp   += qd * Z[i];
      }
      // sum the 4 d-groups (lanes differing in bits 3..4) -> full q.S, q.z
      p  += __shfl_xor(p, 8, 32);  p  += __shfl_xor(p, 16, 32);
      dp += __shfl_xor(dp, 8, 32); dp += __shfl_xor(dp, 16, 32);
      if (dg == 0) {
        const int l = c * CHUNK + s;
        attn[(size_t)(b * LSEQ + l) * EE + h * HDIM + e] = p / fmaxf(dp, 1e-6f);
      }
    }
    __syncthreads();                      // all reads done before overwrite
  }
}

// ---------------------------------------------------------------------------
// Prediction head: gather state tokens (index 3k+1) and project E -> NA=5.
// ---------------------------------------------------------------------------
__global__ void pred_kernel(const float* __restrict__ xf,
                            const float* __restrict__ pred_w,
                            const float* __restrict__ pred_b,
                            float* __restrict__ out) {
  const int rk = blockIdx.x;            // 0 .. B*K-1
  const int b  = rk / KK;
  const int k  = rk % KK;
  const int lane = threadIdx.x;
  const float* xr = xf + (size_t)(b * LSEQ + 3 * k + 1) * EE;
  float acc[NACT] = {0.f, 0.f, 0.f, 0.f, 0.f};
  for (int e = lane; e < EE; e += 32) {
    const float xv = xr[e];
    #pragma unroll
    for (int c = 0; c < NACT; ++c) acc[c] += xv * pred_w[e * NACT + c];
  }
  #pragma unroll
  for (int c = 0; c < NACT; ++c) acc[c] = wave_sum32(acc[c]);
  if (lane == 0) {
    #pragma unroll
    for (int c = 0; c < NACT; ++c) out[rk * NACT + c] = acc[c] + pred_b[c];
  }
}

// ---------------------------------------------------------------------------
extern "C" void kernel_launch(void* const* d_in, const int* in_sizes, int n_in,
                              void* d_out, int out_size, void* d_ws, size_t ws_size,
                              hipStream_t stream) {
  (void)in_sizes; (void)n_in; (void)out_size; (void)ws_size;
  const float* rtg      = (const float*)d_in[0];
  const float* state    = (const float*)d_in[1];
  const float* action   = (const float*)d_in[2];
  const float* rtg_w    = (const float*)d_in[3];
  const float* rtg_b    = (const float*)d_in[4];
  const float* state_w  = (const float*)d_in[5];
  const float* state_b  = (const float*)d_in[6];
  const float* action_w = (const float*)d_in[7];
  const float* action_b = (const float*)d_in[8];
  const float* pos_emb  = (const float*)d_in[9];
  const float* norm1_g  = (const float*)d_in[10];
  const float* norm1_b  = (const float*)d_in[11];
  const float* qkv_w    = (const float*)d_in[12];
  const float* qkv_b    = (const float*)d_in[13];
  const float* out_w    = (const float*)d_in[14];
  const float* out_b    = (const float*)d_in[15];
  const float* norm2_g  = (const float*)d_in[16];
  const float* norm2_b  = (const float*)d_in[17];
  const float* ffn1_w   = (const float*)d_in[18];
  const float* ffn1_b   = (const float*)d_in[19];
  const float* ffn2_w   = (const float*)d_in[20];
  const float* ffn2_b   = (const float*)d_in[21];
  const float* normf_g  = (const float*)d_in[22];
  const float* normf_b  = (const float*)d_in[23];
  const float* pred_w   = (const float*)d_in[24];
  const float* pred_b   = (const float*)d_in[25];

  // workspace carve-out (floats): x, h, qkv, attn, f1  (~31.5 MB total)
  float* x    = (float*)d_ws;
  float* h    = x    + (size_t)MROWS * EE;
  float* qkv  = h    + (size_t)MROWS * EE;
  float* attn = qkv  + (size_t)MROWS * 3 * EE;
  float* f1   = attn + (size_t)MROWS * EE;

  // 1) token embeddings + positional, interleaved (r,s,a)
  embed_kernel<<<MROWS, 256, 0, stream>>>(rtg, state, action,
                                          rtg_w, rtg_b, state_w, state_b,
                                          action_w, action_b, pos_emb, x);

  for (int l = 0; l < NLAYER; ++l) {
    // LN1
    layernorm_kernel<<<MROWS, 32, 0, stream>>>(x, norm1_g + l * EE, norm1_b + l * EE, h);
    // qkv projection: (3072 x 256) @ (256 x 768), 16x128 strips
    {
      const int waves = (MROWS / 16) * ((3 * EE) / 128);
      gemm_wmma_f32<0, false, 8><<<waves / 8, 256, 0, stream>>>(
          h, qkv_w + (size_t)l * EE * 3 * EE, qkv_b + l * 3 * EE,
          nullptr, qkv, MROWS, EE, 3 * EE);
    }
    // causal linear-attention scan (8 blocks: B*H)
    linear_attn_scan<<<BB * NHEAD, 256, 0, stream>>>(qkv, attn);
    // out projection + residual: x = x + attn @ out_w + out_b, 16x64 strips
    {
      const int waves = (MROWS / 16) * (EE / 64);
      gemm_wmma_f32<0, true, 4><<<waves / 8, 256, 0, stream>>>(
          attn, out_w + (size_t)l * EE * EE, out_b + l * EE,
          x, x, MROWS, EE, EE);
    }
    // LN2
    layernorm_kernel<<<MROWS, 32, 0, stream>>>(x, norm2_g + l * EE, norm2_b + l * EE, h);
    // FFN1 + exact GELU: (3072 x 256) @ (256 x 1024), 16x128 strips
    {
      const int waves = (MROWS / 16) * (FFD / 128);
      gemm_wmma_f32<1, false, 8><<<waves / 8, 256, 0, stream>>>(
          h, ffn1_w + (size_t)l * EE * FFD, ffn1_b + l * FFD,
          nullptr, f1, MROWS, EE, FFD);
    }
    // FFN2 + residual: x = x + f1 @ ffn2_w + ffn2_b, 16x64 strips
    {
      const int waves = (MROWS / 16) * (EE / 64);
      gemm_wmma_f32<0, true, 4><<<waves / 8, 256, 0, stream>>>(
          f1, ffn2_w + (size_t)l * FFD * EE, ffn2_b + l * EE,
          x, x, MROWS, FFD, EE);
    }
  }

  // final LN, then predict actions from state tokens
  layernorm_kernel<<<MROWS, 32, 0, stream>>>(x, normf_g, normf_b, h);
  pred_kernel<<<BB * KK, 32, 0, stream>>>(h, pred_w, pred_b, (float*)d_out);
}